// PointerLSTM_90263032692766
// MI455X (gfx1250) — compile-verified
//
#include <hip/hip_runtime.h>
#include <hip/hip_bf16.h>

// Sizes fixed by the reference
#define BB 64
#define LL 1024
#define DD 256
#define HH 256
#define Z4 1024          // 4*H
#define TB 16            // timestep block in attention
#define LDST 260         // padded LDS stride (260 % 64 == 4 -> conflict-free)

typedef __attribute__((ext_vector_type(2))) float v2f;
typedef __attribute__((ext_vector_type(8))) float v8f;

__device__ __forceinline__ v8f wmma4(v2f a, v2f b, v8f c) {
  // D = A(16x4,f32) x B(4x16,f32) + C(16x16,f32)
  return __builtin_amdgcn_wmma_f32_16x16x4_f32(false, a, false, b, (short)0, c,
                                               false, false);
}

__device__ __forceinline__ float fast_tanh(float x) {
#if defined(__AMDGCN__) && __has_builtin(__builtin_amdgcn_tanhf)
  return __builtin_amdgcn_tanhf(x);
#else
  float e = __expf(2.0f * x);
  return 1.0f - 2.0f / (e + 1.0f);
#endif
}

__device__ __forceinline__ float fast_sigmoid(float x) {
  return 1.0f / (1.0f + __expf(-x));
}

// ---------------------------------------------------------------------------
// xz = x[:, L-1, :] @ Wk + b        -> (64, 1024)
// ---------------------------------------------------------------------------
__global__ __launch_bounds__(256) void xz_kernel(
    const float* __restrict__ x, const float* __restrict__ Wk,
    const float* __restrict__ bvec, float* __restrict__ xz) {
  int idx = blockIdx.x * 256 + threadIdx.x;  // 0..65535
  int b = idx >> 10;
  int n = idx & (Z4 - 1);
  const float* xr = x + ((size_t)b * LL + (LL - 1)) * DD;
  float acc = bvec[n];
  for (int d = 0; d < DD; ++d) acc += xr[d] * Wk[(size_t)d * Z4 + n];
  xz[idx] = acc;
}

// ---------------------------------------------------------------------------
// w1t[b][h][l] = (x@W1 + b1) transposed, via f32 WMMA.
// Orientation: C[M=h, N=l] = sum_d W1[d][hbase+M] * x[b][lbase+N][d]
// A loads (W1) and C stores are lane-coalesced; B (x) streams rows over k.
// One 16x16 tile per wave; 65536 tiles -> 8192 blocks of 8 waves.
// ---------------------------------------------------------------------------
__global__ __launch_bounds__(256) void w1t_gemm(
    const float* __restrict__ x, const float* __restrict__ W1,
    const float* __restrict__ b1, float* __restrict__ w1t) {
  int wave = threadIdx.x >> 5;
  int lane = threadIdx.x & 31;
  int col = lane & 15;
  int hf = lane >> 4;      // lane half
  int Mofs = hf * 8;       // C row offset for this half
  int kofs = hf * 2;       // K offset for A/B operands

  int tile = blockIdx.x * 8 + wave;  // b(6b) | ht(4b) | lt(6b)
  int b = tile >> 10;
  int ht = (tile >> 6) & 15;
  int lt = tile & 63;
  int hbase = ht * 16, lbase = lt * 16;

  const float* xrow = x + ((size_t)b * LL + lbase + col) * DD;

  v8f acc;
#pragma unroll
  for (int r = 0; r < 8; ++r) acc[r] = b1[hbase + r + Mofs];

  for (int k = 0; k < DD; k += 4) {
    v2f a;
    a.x = W1[(size_t)(k + kofs) * HH + hbase + col];
    a.y = W1[(size_t)(k + kofs + 1) * HH + hbase + col];
    v2f bb = *(const v2f*)&xrow[k + kofs];
    acc = wmma4(a, bb, acc);
  }

#pragma unroll
  for (int r = 0; r < 8; ++r)
    w1t[((size_t)b * HH + hbase + r + Mofs) * LL + lbase + col] = acc[r];
}

// ---------------------------------------------------------------------------
// Persistent LSTM recurrence. One WG per 16-batch group (grid = 4).
// 16 waves; wave w owns h columns [16w, 16w+16) and the four z gate tiles
// at columns {16w, 256+16w, 512+16w, 768+16w} == (i,f,g,o) for that slice.
// h lives in LDS (16 x 260 padded, conflict-free ds_load_b64 A-operand).
// Also emits w2_all[t][b][:] = h@W2 + b2 each step (fused WMMA).
// ---------------------------------------------------------------------------
__global__ __launch_bounds__(512) void lstm_recur(
    const float* __restrict__ xz, const float* __restrict__ Wr,
    const float* __restrict__ W2, const float* __restrict__ b2,
    float* __restrict__ w2_all) {
  __shared__ float hlds[16 * LDST];

  int wave = threadIdx.x >> 5;   // 0..15
  int lane = threadIdx.x & 31;
  int col = lane & 15;
  int hf = lane >> 4;
  int Mofs = hf * 8;
  int kofs = hf * 2;
  int bbase = blockIdx.x * 16;

  for (int i = threadIdx.x; i < 16 * LDST; i += 512) hlds[i] = 0.0f;

  // xz tiles (constant across steps) = initial C for the z accumulation
  v8f zinit[4];
#pragma unroll
  for (int g = 0; g < 4; ++g) {
    int n0 = g * HH + wave * 16 + col;
#pragma unroll
    for (int r = 0; r < 8; ++r)
      zinit[g][r] = xz[(size_t)(bbase + r + Mofs) * Z4 + n0];
  }

  v8f creg = {0, 0, 0, 0, 0, 0, 0, 0};
  const float b2v = b2[wave * 16 + col];
  const int aoff = (lane & 15) * LDST + kofs;  // A-operand LDS base

  __syncthreads();

  for (int t = 0; t < LL; ++t) {
    // ---- z = xz + h @ Wr (4 gate tiles) ----
    v8f z0 = zinit[0], z1 = zinit[1], z2 = zinit[2], z3 = zinit[3];
    for (int k = 0; k < HH; k += 4) {
      v2f a = *(const v2f*)&hlds[aoff + k];
      int rowk = (k + kofs) * Z4;
      int n0 = wave * 16 + col;
      v2f b0, b1v, b2t, b3;
      b0.x = Wr[rowk + n0];               b0.y = Wr[rowk + Z4 + n0];
      b1v.x = Wr[rowk + (HH) + n0];       b1v.y = Wr[rowk + Z4 + HH + n0];
      b2t.x = Wr[rowk + (2 * HH) + n0];   b2t.y = Wr[rowk + Z4 + 2 * HH + n0];
      b3.x = Wr[rowk + (3 * HH) + n0];    b3.y = Wr[rowk + Z4 + 3 * HH + n0];
      z0 = wmma4(a, b0, z0);
      z1 = wmma4(a, b1v, z1);
      z2 = wmma4(a, b2t, z2);
      z3 = wmma4(a, b3, z3);
    }

    // ---- gates / state update (elementwise, tile-local) ----
    v8f hreg;
#pragma unroll
    for (int r = 0; r < 8; ++r) {
      float iv = fast_sigmoid(z0[r]);
      float fv = fast_sigmoid(z1[r]);
      float gv = fast_tanh(z2[r]);
      float ov = fast_sigmoid(z3[r]);
      float cv = fv * creg[r] + iv * gv;
      creg[r] = cv;
      hreg[r] = ov * fast_tanh(cv);
    }

    __syncthreads();  // all waves done reading old h
#pragma unroll
    for (int r = 0; r < 8; ++r)
      hlds[(r + Mofs) * LDST + wave * 16 + col] = hreg[r];
    __syncthreads();  // new h visible

    // ---- w2_d = h @ W2 + b2 (this wave's 16 columns) ----
    v8f acc = {0, 0, 0, 0, 0, 0, 0, 0};
    for (int k = 0; k < HH; k += 4) {
      v2f a = *(const v2f*)&hlds[aoff + k];
      v2f bb;
      int n0 = wave * 16 + col;
      bb.x = W2[(k + kofs) * HH + n0];
      bb.y = W2[(k + kofs + 1) * HH + n0];
      acc = wmma4(a, bb, acc);
    }
#pragma unroll
    for (int r = 0; r < 8; ++r)
      w2_all[((size_t)t * BB + bbase + r + Mofs) * HH + wave * 16 + col] =
          acc[r] + b2v;
  }
}

// ---------------------------------------------------------------------------
// Attention: one WG per (b, 16-timestep block). Thread tid owns l = tid+256j.
// scores[l,t] = sum_h tanh(w1t[b][h][l] + w2[t][h]) * V[h]; softmax over l.
// (bV is constant across l -> softmax-invariant -> omitted.)
// ---------------------------------------------------------------------------
__global__ __launch_bounds__(256) void attn_kernel(
    const float* __restrict__ w1t, const float* __restrict__ w2_all,
    const float* __restrict__ V, float* __restrict__ out) {
  __shared__ float w2s[TB * HH];
  __shared__ float vs[HH];
  __shared__ float red[TB * 256];

  int b = blockIdx.x >> 6;
  int tb = (blockIdx.x & 63) * TB;
  int tid = threadIdx.x;

  for (int i = tid; i < TB * HH; i += 256) {
    int t = i >> 8, h = i & (HH - 1);
    w2s[i] = w2_all[((size_t)(tb + t) * BB + b) * HH + h];
  }
  vs[tid] = V[tid];
  __syncthreads();

  float acc[4][TB];
#pragma unroll
  for (int j = 0; j < 4; ++j)
#pragma unroll
    for (int t = 0; t < TB; ++t) acc[j][t] = 0.0f;

  const float* w1b = w1t + (size_t)b * HH * LL;
  for (int h = 0; h < HH; ++h) {
    float vh = vs[h];
    const float* row = w1b + (size_t)h * LL;
    float a0 = row[tid];
    float a1 = row[tid + 256];
    float a2 = row[tid + 512];
    float a3 = row[tid + 768];
#pragma unroll
    for (int t = 0; t < TB; ++t) {
      float w = w2s[t * HH + h];
      acc[0][t] += fast_tanh(a0 + w) * vh;
      acc[1][t] += fast_tanh(a1 + w) * vh;
      acc[2][t] += fast_tanh(a2 + w) * vh;
      acc[3][t] += fast_tanh(a3 + w) * vh;
    }
  }

  // softmax over l (1024 = 4 per thread x 256 threads), all TB rows at once
#pragma unroll
  for (int t = 0; t < TB; ++t) {
    float m = fmaxf(fmaxf(acc[0][t], acc[1][t]), fmaxf(acc[2][t], acc[3][t]));
    red[t * 256 + tid] = m;
  }
  __syncthreads();
  for (int s = 128; s > 0; s >>= 1) {
    if (tid < s) {
#pragma unroll
      for (int t = 0; t < TB; ++t)
        red[t * 256 + tid] = fmaxf(red[t * 256 + tid], red[t * 256 + tid + s]);
    }
    __syncthreads();
  }
  float mt[TB];
#pragma unroll
  for (int t = 0; t < TB; ++t) mt[t] = red[t * 256];
  __syncthreads();

#pragma unroll
  for (int t = 0; t < TB; ++t) {
    float s0 = 0.0f;
#pragma unroll
    for (int j = 0; j < 4; ++j) {
      acc[j][t] = __expf(acc[j][t] - mt[t]);
      s0 += acc[j][t];
    }
    red[t * 256 + tid] = s0;
  }
  __syncthreads();
  for (int s = 128; s > 0; s >>= 1) {
    if (tid < s) {
#pragma unroll
      for (int t = 0; t < TB; ++t)
        red[t * 256 + tid] += red[t * 256 + tid + s];
    }
    __syncthreads();
  }

#pragma unroll
  for (int t = 0; t < TB; ++t) {
    float inv = 1.0f / red[t * 256];
    size_t base = ((size_t)b * LL + tb + t) * LL;
#pragma unroll
    for (int j = 0; j < 4; ++j) out[base + tid + j * 256] = acc[j][t] * inv;
  }
}

// ---------------------------------------------------------------------------
extern "C" void kernel_launch(void* const* d_in, const int* in_sizes, int n_in,
                              void* d_out, int out_size, void* d_ws,
                              size_t ws_size, hipStream_t stream) {
  const float* x = (const float*)d_in[0];    // (64,1024,256)
  const float* Wk = (const float*)d_in[1];   // (256,1024)
  const float* Wr = (const float*)d_in[2];   // (256,1024)
  const float* bz = (const float*)d_in[3];   // (1024)
  const float* W1 = (const float*)d_in[4];   // (256,256)
  const float* b1 = (const float*)d_in[5];   // (256)
  const float* W2 = (const float*)d_in[6];   // (256,256)
  const float* b2 = (const float*)d_in[7];   // (256)
  const float* V = (const float*)d_in[8];    // (256,1)
  // d_in[9] = bV: softmax-invariant, unused
  float* out = (float*)d_out;                // (64,1024,1024)

  float* w1t = (float*)d_ws;                        // 64 MB  (B,H,L)
  float* w2_all = w1t + (size_t)BB * HH * LL;       // 64 MB  (T,B,H)
  float* xz = w2_all + (size_t)LL * BB * HH;        // 256 KB (B,4H)

  xz_kernel<<<256, 256, 0, stream>>>(x, Wk, bz, xz);
  w1t_gemm<<<8192, 256, 0, stream>>>(x, W1, b1, w1t);
  lstm_recur<<<4, 512, 0, stream>>>(xz, Wr, W2, b2, w2_all);
  attn_kernel<<<BB * (LL / TB), 256, 0, stream>>>(w1t, w2_all, V, out);
}